// SeSaMe_86406152061027
// MI455X (gfx1250) — compile-verified
//
#include <hip/hip_runtime.h>
#include <hip/hip_bf16.h>

typedef float v2f __attribute__((ext_vector_type(2)));
typedef float v8f __attribute__((ext_vector_type(8)));
typedef int   v4i __attribute__((vector_size(16)));   // matches builtin's pointee type

#if __has_builtin(__builtin_amdgcn_global_load_async_to_lds_b128) && \
    __has_builtin(__builtin_amdgcn_s_wait_asynccnt)
#define HAS_ASYNC 1
#else
#define HAS_ASYNC 0
#endif

__device__ __forceinline__ float dev_gelu(float v){ return 0.5f*v*(1.f+erff(v*0.70710678118654752f)); }
__device__ __forceinline__ float dev_silu(float v){ return v/(1.f+__expf(-v)); }
__device__ __forceinline__ float dev_softplus(float v){ return v>20.f?v:log1pf(__expf(v)); }

#if HAS_ASYNC
__device__ __forceinline__ void async_load_b128(const float* g, float* l){
  __builtin_amdgcn_global_load_async_to_lds_b128(
      (__attribute__((address_space(1))) v4i*)g,
      (__attribute__((address_space(3))) v4i*)l, 0, 0);
}
#endif

// One V_WMMA_F32_16X16X4_F32 step: C += X(16xKp)[:,k0:k0+4] @ Wp(16xKp)[:,k0:k0+4]^T
// X, Wp in LDS, zero-padded. ISA layout: lane l -> row m=l&15, K-pair kb=(l>>4)*2.
__device__ __forceinline__ v8f wmma_k4(v8f c, const float* X, int ldx,
                                       const float* Wp, int ldw, int k0){
  int l  = threadIdx.x & 31;
  int m  = l & 15;
  int kb = (l >> 4) << 1;
  v2f a, b;
  a.x = X[m*ldx + k0 + kb];  a.y = X[m*ldx + k0 + kb + 1];
  b.x = Wp[m*ldw + k0 + kb]; b.y = Wp[m*ldw + k0 + kb + 1];
  return __builtin_amdgcn_wmma_f32_16x16x4_f32(false, a, false, b, (short)0, c, false, false);
}

// ---------------- front end: x -> x2 (l1+gelu) -> x3 (l2+gelu) ------------
__global__ __launch_bounds__(32)
void k_front(const float* __restrict__ x,
             const float* __restrict__ l1w, const float* __restrict__ l1b,
             const float* __restrict__ l2w, const float* __restrict__ l2b,
             float* __restrict__ x2out, float* __restrict__ x3out){
  __shared__ float sh1[256];   // 64 tokens x 4
  __shared__ float sh2[128];   // 64 tokens x 2 == 16 tokens x 8
  __shared__ float w1[64];     // 16x4 padded (l1_w is 2x4)
  __shared__ float w2[128];    // 16x8 padded (l2_w is 4x8)
  int l   = threadIdx.x;
  int blk = blockIdx.x;
  int b   = blk >> 8;          // 256 blocks per batch
  int part= blk & 255;
  for (int i=l;i<64;i+=32)  w1[i]=0.f;
  for (int i=l;i<128;i+=32) w2[i]=0.f;
  __syncthreads();
  for (int i=l;i<8;i+=32)  w1[i]=l1w[i];
  for (int i=l;i<32;i+=32) w2[i]=l2w[i];
  const float* xb = x + ((size_t)b<<16) + ((size_t)part<<8);
  for (int i=l;i<256;i+=32) sh1[i]=xb[i];
  __syncthreads();
  int n = l & 15, mh = (l>>4)<<3;
  for (int t=0;t<4;t++){
    v8f c = {};
    c = wmma_k4(c, &sh1[t*64], 4, w1, 4, 0);
    if (n < 2){
      float bb = l1b[n];
      #pragma unroll
      for (int r=0;r<8;r++) sh2[(t*16 + r + mh)*2 + n] = dev_gelu(c[r] + bb);
    }
    __syncthreads();
  }
  { // emit x2
    size_t base = ((size_t)b*16384 + (size_t)part*64)*2;
    for (int i=l;i<128;i+=32) x2out[base+i] = sh2[i];
  }
  v8f c = {};
  c = wmma_k4(c, sh2, 8, w2, 8, 0);
  c = wmma_k4(c, sh2, 8, w2, 8, 4);
  if (n < 4){
    float bb = l2b[n];
    size_t base = ((size_t)b*4096 + (size_t)part*16)*4;
    #pragma unroll
    for (int r=0;r<8;r++) x3out[base + (size_t)(r+mh)*4 + n] = dev_gelu(c[r] + bb);
  }
}

// ---------------- generic tiny linear + gelu ------------------------------
template<int K, int N>
__global__ __launch_bounds__(32)
void k_lin(const float* __restrict__ in, const float* __restrict__ W,
           const float* __restrict__ bias, float* __restrict__ out){
  constexpr int Kp = (K + 3) & ~3;
  __shared__ float sh [16*Kp];
  __shared__ float wsh[16*Kp];
  int l = threadIdx.x;
  for (int i=l;i<16*Kp;i+=32){ sh[i]=0.f; wsh[i]=0.f; }
  __syncthreads();
  size_t tok0 = (size_t)blockIdx.x*16;
  for (int i=l;i<16*K;i+=32) sh [(i/K)*Kp + (i%K)] = in[tok0*K + i];
  for (int i=l;i<N*K; i+=32) wsh[(i/K)*Kp + (i%K)] = W[i];
  __syncthreads();
  v8f c = {};
  #pragma unroll
  for (int k0=0;k0<Kp;k0+=4) c = wmma_k4(c, sh, Kp, wsh, Kp, k0);
  int n = l & 15, mh = (l>>4)<<3;
  if (n < N){
    float bb = bias[n];
    #pragma unroll
    for (int r=0;r<8;r++) out[(tok0 + r + mh)*N + n] = dev_gelu(c[r] + bb);
  }
}

// ---------------- mamba in-projection: xz = u @ Wi^T, split x/z ----------
template<int DM, int DI>
__global__ __launch_bounds__(32)
void k_inproj(const float* __restrict__ u, const float* __restrict__ Wi,
              float* __restrict__ xout, float* __restrict__ zout){
  __shared__ float sh [64];
  __shared__ float wsh[64];
  int l = threadIdx.x;
  for (int i=l;i<64;i+=32){ sh[i]=0.f; wsh[i]=0.f; }
  __syncthreads();
  size_t tok0 = (size_t)blockIdx.x*16;
  for (int i=l;i<16*DM;  i+=32) sh [(i/DM)*4 + (i%DM)] = u[tok0*DM + i];
  for (int i=l;i<2*DI*DM;i+=32) wsh[(i/DM)*4 + (i%DM)] = Wi[i];
  __syncthreads();
  v8f c = {};
  c = wmma_k4(c, sh, 4, wsh, 4, 0);
  int n = l & 15, mh = (l>>4)<<3;
  #pragma unroll
  for (int r=0;r<8;r++){
    int m = r + mh; float v = c[r];
    if      (n < DI)   xout[(tok0+m)*DI + n]        = v;
    else if (n < 2*DI) zout[(tok0+m)*DI + (n - DI)] = v;
  }
}

// ---------------- causal depthwise conv(K=4) + SiLU ----------------------
template<int DI>
__global__ void k_conv(const float* __restrict__ xin, const float* __restrict__ cw,
                       const float* __restrict__ cb,  float* __restrict__ xout,
                       int T, int n){
  int i = blockIdx.x*blockDim.x + threadIdx.x;
  if (i >= n) return;
  int c = i % DI;
  int bt = i / DI;
  int t = bt % T;
  int b = bt / T;
  const float* base = xin + (size_t)b*T*DI;
  float acc = cb[c];
  #pragma unroll
  for (int k=0;k<4;k++){
    int tt = t - 3 + k;
    if (tt >= 0) acc += base[(size_t)tt*DI + c]*cw[c*4+k];
  }
  xout[i] = dev_silu(acc);
}

// ---------------- selective scan, phase 1: per-chunk local scan ----------
// lane l owns E=DI*16/32 state elements e=l*E+j : d=e/16 (constant per lane), s=e%16
template<int DI, int CHUNK>
__global__ __launch_bounds__(32)
void k_scan1(const float* __restrict__ xc, const float* __restrict__ Wx,
             const float* __restrict__ Wdt, const float* __restrict__ bdt,
             const float* __restrict__ Alog, int T, int nchunks,
             float* __restrict__ apout, float* __restrict__ heout){
  constexpr int E = (DI*16)/32;
  int l = threadIdx.x;
  int b = blockIdx.x / nchunks, ch = blockIdx.x % nchunks;
  int d = (l*E) >> 4, sbase = (l*E) & 15;
  float Av[E], h[E], ap[E], wB[E][DI], w0[DI];
  #pragma unroll
  for (int j=0;j<E;j++){
    Av[j] = -__expf(Alog[d*16 + sbase + j]);
    h[j] = 0.f; ap[j] = 1.f;
    #pragma unroll
    for (int c=0;c<DI;c++) wB[j][c] = Wx[(1 + sbase + j)*DI + c];
  }
  #pragma unroll
  for (int c=0;c<DI;c++) w0[c] = Wx[c];
  float wdtd = Wdt[d], bdtd = bdt[d];
  const float* xb = xc + ((size_t)b*T + (size_t)ch*CHUNK)*DI;

  auto step = [&](const float* xv){
    float dtraw = 0.f;
    #pragma unroll
    for (int c=0;c<DI;c++) dtraw += xv[c]*w0[c];
    float dt  = dev_softplus(dtraw*wdtd + bdtd);
    float dtx = dt * xv[d];
    #pragma unroll
    for (int j=0;j<E;j++){
      float Bs = 0.f;
      #pragma unroll
      for (int c=0;c<DI;c++) Bs += xv[c]*wB[j][c];
      float a = __expf(dt*Av[j]);
      h[j]  = a*h[j] + dtx*Bs;
      ap[j]*= a;
    }
  };

#if HAS_ASYNC
  constexpr int TT = 256/DI;        // tokens per 1KB tile
  constexpr int NTILE = CHUNK/TT;
  __shared__ float xsh[2][256];
  async_load_b128(xb + l*4,       &xsh[0][l*4]);
  async_load_b128(xb + 128 + l*4, &xsh[0][128 + l*4]);
  for (int tile=0; tile<NTILE; ++tile){
    if (tile+1 < NTILE){
      const float* s = xb + (size_t)(tile+1)*256 + l*4;
      async_load_b128(s,       &xsh[(tile+1)&1][l*4]);
      async_load_b128(s + 128, &xsh[(tile+1)&1][128 + l*4]);
      __builtin_amdgcn_s_wait_asynccnt(2);
    } else {
      __builtin_amdgcn_s_wait_asynccnt(0);
    }
    asm volatile("" ::: "memory");
    const float* xt = &xsh[tile&1][0];
    for (int tt=0; tt<TT; ++tt) step(xt + tt*DI);
    asm volatile("" ::: "memory");
  }
#else
  for (int t=0;t<CHUNK;t++) step(xb + (size_t)t*DI);
#endif
  size_t base = ((size_t)(b*nchunks + ch))*(DI*16) + (size_t)l*E;
  #pragma unroll
  for (int j=0;j<E;j++){ apout[base+j]=ap[j]; heout[base+j]=h[j]; }
}

// ---------------- phase 2: cross-chunk exclusive combine -----------------
template<int DI>
__global__ void k_scan2(const float* __restrict__ ap, const float* __restrict__ he,
                        float* __restrict__ hin, int nchunks, int nelem){
  int i = blockIdx.x*blockDim.x + threadIdx.x;
  if (i >= nelem) return;
  int e = i % (DI*16), b = i / (DI*16);
  float h = 0.f;
  for (int c=0;c<nchunks;c++){
    size_t idx = ((size_t)(b*nchunks + c))*(DI*16) + e;
    hin[idx] = h;
    h = ap[idx]*h + he[idx];
  }
}

// ---------------- phase 3: replay chunk from hin, produce gelu(output) ---
template<int DI, int CHUNK, int DM>
__global__ __launch_bounds__(32)
void k_scan3(const float* __restrict__ xc, const float* __restrict__ z,
             const float* __restrict__ Wx, const float* __restrict__ Wdt,
             const float* __restrict__ bdt, const float* __restrict__ Alog,
             const float* __restrict__ Dp,  const float* __restrict__ Wo,
             const float* __restrict__ hin, int T, int nchunks,
             float* __restrict__ out){
  constexpr int E = (DI*16)/32;
  constexpr int G = 32/DI;               // lanes sharing one d
  int l = threadIdx.x;
  int b = blockIdx.x / nchunks, ch = blockIdx.x % nchunks;
  int d = (l*E) >> 4, sbase = (l*E) & 15;
  float Av[E], h[E], wB[E][DI], wC[E][DI], w0[DI], wor[DI];
  size_t hbase = ((size_t)(b*nchunks + ch))*(DI*16) + (size_t)l*E;
  #pragma unroll
  for (int j=0;j<E;j++){
    Av[j] = -__expf(Alog[d*16 + sbase + j]);
    h[j]  = hin[hbase + j];
    #pragma unroll
    for (int c=0;c<DI;c++){
      wB[j][c] = Wx[(1  + sbase + j)*DI + c];
      wC[j][c] = Wx[(17 + sbase + j)*DI + c];
    }
  }
  #pragma unroll
  for (int c=0;c<DI;c++) w0[c] = Wx[c];
  int orow = (l < DM) ? l : 0;
  #pragma unroll
  for (int c=0;c<DI;c++) wor[c] = Wo[orow*DI + c];
  float wdtd = Wdt[d], bdtd = bdt[d], Dd = Dp[d];
  const float* xb = xc + ((size_t)b*T + (size_t)ch*CHUNK)*DI;
  const float* zb = z  + ((size_t)b*T + (size_t)ch*CHUNK)*DI;
  float* ob       = out + ((size_t)b*T + (size_t)ch*CHUNK)*DM;

  auto step = [&](const float* xv, int t){
    float dtraw = 0.f;
    #pragma unroll
    for (int c=0;c<DI;c++) dtraw += xv[c]*w0[c];
    float dt  = dev_softplus(dtraw*wdtd + bdtd);
    float dtx = dt * xv[d];
    float ps = 0.f;
    #pragma unroll
    for (int j=0;j<E;j++){
      float Bs = 0.f, Cs = 0.f;
      #pragma unroll
      for (int c=0;c<DI;c++){ Bs += xv[c]*wB[j][c]; Cs += xv[c]*wC[j][c]; }
      float a = __expf(dt*Av[j]);
      h[j] = a*h[j] + dtx*Bs;
      ps  += h[j]*Cs;
    }
    #pragma unroll
    for (int msk = G>>1; msk >= 1; msk >>= 1) ps += __shfl_xor(ps, msk, 32);
    float yd = ps + xv[d]*Dd;              // all lanes in a d-group hold y[d]
    float o = 0.f;
    #pragma unroll
    for (int dd=0; dd<DI; dd++){
      float ydd = __shfl(yd, dd*G, 32);
      float zv  = zb[(size_t)t*DI + dd];
      o += ydd * dev_silu(zv) * wor[dd];
    }
    if (l < DM) ob[(size_t)t*DM + l] = dev_gelu(o);
  };

#if HAS_ASYNC
  constexpr int TT = 256/DI;        // tokens per 1KB tile
  constexpr int NTILE = CHUNK/TT;
  __shared__ float xsh[2][256];
  async_load_b128(xb + l*4,       &xsh[0][l*4]);
  async_load_b128(xb + 128 + l*4, &xsh[0][128 + l*4]);
  for (int tile=0; tile<NTILE; ++tile){
    if (tile+1 < NTILE){
      const float* s = xb + (size_t)(tile+1)*256 + l*4;
      async_load_b128(s,       &xsh[(tile+1)&1][l*4]);
      async_load_b128(s + 128, &xsh[(tile+1)&1][128 + l*4]);
      __builtin_amdgcn_s_wait_asynccnt(2);
    } else {
      __builtin_amdgcn_s_wait_asynccnt(0);
    }
    asm volatile("" ::: "memory");
    const float* xt = &xsh[tile&1][0];
    for (int tt=0; tt<TT; ++tt) step(xt + tt*DI, tile*TT + tt);
    asm volatile("" ::: "memory");
  }
#else
  for (int t=0;t<CHUNK;t++) step(xb + (size_t)t*DI, t);
#endif
}

// ---------------- elementwise add (flat reshapes are layout-preserving) --
__global__ void k_add(const float* __restrict__ a, const float* __restrict__ b,
                      float* __restrict__ o, int n){
  int i = blockIdx.x*blockDim.x + threadIdx.x;
  if (i < n) o[i] = a[i] + b[i];
}

extern "C" void kernel_launch(void* const* d_in, const int* in_sizes, int n_in,
                              void* d_out, int out_size, void* d_ws, size_t ws_size,
                              hipStream_t stream){
  (void)in_sizes; (void)n_in; (void)out_size; (void)ws_size;
  const float* x   = (const float*)d_in[0];
  const float* l1w = (const float*)d_in[1]; const float* l1b = (const float*)d_in[2];
  const float* l2w = (const float*)d_in[3]; const float* l2b = (const float*)d_in[4];
  const float* l3w = (const float*)d_in[5]; const float* l3b = (const float*)d_in[6];
  const float* l4w = (const float*)d_in[7]; const float* l4b = (const float*)d_in[8];
  const float *Wi[3], *Cw[3], *Cb[3], *Wx[3], *Wdt[3], *Bdt[3], *Alog[3], *Dp[3], *Wo[3];
  for (int k=0;k<3;k++){
    int o = 9 + 9*k;
    Wi[k]=(const float*)d_in[o+0]; Cw[k]=(const float*)d_in[o+1]; Cb[k]=(const float*)d_in[o+2];
    Wx[k]=(const float*)d_in[o+3]; Wdt[k]=(const float*)d_in[o+4]; Bdt[k]=(const float*)d_in[o+5];
    Alog[k]=(const float*)d_in[o+6]; Dp[k]=(const float*)d_in[o+7]; Wo[k]=(const float*)d_in[o+8];
  }
  float* yout = (float*)d_out;

  const int Bb = 16, NCH = 64;
  const int T1 = 4096, T2 = 16384, T3 = 65536;

  float* w = (float*)d_ws; size_t off = 0;
  auto alloc = [&](size_t n){ float* p = w + off; off += n; return p; };
  float* x2  = alloc((size_t)Bb*T2*2);
  float* x3  = alloc((size_t)Bb*T1*4);
  float* xm1 = alloc((size_t)Bb*T1*8); float* zm1 = alloc((size_t)Bb*T1*8); float* xc1 = alloc((size_t)Bb*T1*8);
  float* ap1 = alloc((size_t)Bb*NCH*128); float* he1 = alloc((size_t)Bb*NCH*128); float* hi1 = alloc((size_t)Bb*NCH*128);
  float* g1  = alloc((size_t)Bb*T1*4);
  float* x4  = alloc((size_t)Bb*T1*4);
  float* x5  = alloc((size_t)Bb*T1*8);
  float* u2  = alloc((size_t)Bb*T2*2);
  float* xm2 = alloc((size_t)Bb*T2*4); float* zm2 = alloc((size_t)Bb*T2*4); float* xc2 = alloc((size_t)Bb*T2*4);
  float* ap2 = alloc((size_t)Bb*NCH*64); float* he2 = alloc((size_t)Bb*NCH*64); float* hi2 = alloc((size_t)Bb*NCH*64);
  float* x6  = alloc((size_t)Bb*T2*2);
  float* x7  = alloc((size_t)Bb*T2*4);
  float* u3  = alloc((size_t)Bb*T3);
  float* xm3 = alloc((size_t)Bb*T3*2); float* zm3 = alloc((size_t)Bb*T3*2); float* xc3 = alloc((size_t)Bb*T3*2);
  float* ap3 = alloc((size_t)Bb*NCH*32); float* he3 = alloc((size_t)Bb*NCH*32); float* hi3 = alloc((size_t)Bb*NCH*32);

  // ---- front end ----
  k_front<<<Bb*256, 32, 0, stream>>>(x, l1w, l1b, l2w, l2b, x2, x3);

  // ---- mamba 1 (dm=4, di=8, T=4096) ----
  k_inproj<4,8><<<(Bb*T1)/16, 32, 0, stream>>>(x3, Wi[0], xm1, zm1);
  { int n=Bb*T1*8; k_conv<8><<<(n+255)/256, 256, 0, stream>>>(xm1, Cw[0], Cb[0], xc1, T1, n); }
  k_scan1<8,64><<<Bb*NCH, 32, 0, stream>>>(xc1, Wx[0], Wdt[0], Bdt[0], Alog[0], T1, NCH, ap1, he1);
  { int n=Bb*128; k_scan2<8><<<(n+255)/256, 256, 0, stream>>>(ap1, he1, hi1, NCH, n); }
  k_scan3<8,64,4><<<Bb*NCH, 32, 0, stream>>>(xc1, zm1, Wx[0], Wdt[0], Bdt[0], Alog[0], Dp[0], Wo[0], hi1, T1, NCH, g1);
  { int n=Bb*T1*4; k_add<<<(n+255)/256, 256, 0, stream>>>(x3, g1, x4, n); }

  // ---- l3 + residual-with-x2 ----
  k_lin<4,8><<<(Bb*T1)/16, 32, 0, stream>>>(x4, l3w, l3b, x5);
  { int n=Bb*T2*2; k_add<<<(n+255)/256, 256, 0, stream>>>(x5, x2, u2, n); }

  // ---- mamba 2 (dm=2, di=4, T=16384) ----
  k_inproj<2,4><<<(Bb*T2)/16, 32, 0, stream>>>(u2, Wi[1], xm2, zm2);
  { int n=Bb*T2*4; k_conv<4><<<(n+255)/256, 256, 0, stream>>>(xm2, Cw[1], Cb[1], xc2, T2, n); }
  k_scan1<4,256><<<Bb*NCH, 32, 0, stream>>>(xc2, Wx[1], Wdt[1], Bdt[1], Alog[1], T2, NCH, ap2, he2);
  { int n=Bb*64; k_scan2<4><<<(n+255)/256, 256, 0, stream>>>(ap2, he2, hi2, NCH, n); }
  k_scan3<4,256,2><<<Bb*NCH, 32, 0, stream>>>(xc2, zm2, Wx[1], Wdt[1], Bdt[1], Alog[1], Dp[1], Wo[1], hi2, T2, NCH, x6);

  // ---- l4 + residual-with-x1 ----
  k_lin<2,4><<<(Bb*T2)/16, 32, 0, stream>>>(x6, l4w, l4b, x7);
  { int n=Bb*T3; k_add<<<(n+255)/256, 256, 0, stream>>>(x7, x, u3, n); }

  // ---- mamba 3 (dm=1, di=2, T=65536) -> final output ----
  k_inproj<1,2><<<(Bb*T3)/16, 32, 0, stream>>>(u3, Wi[2], xm3, zm3);
  { int n=Bb*T3*2; k_conv<2><<<(n+255)/256, 256, 0, stream>>>(xm3, Cw[2], Cb[2], xc3, T3, n); }
  k_scan1<2,1024><<<Bb*NCH, 32, 0, stream>>>(xc3, Wx[2], Wdt[2], Bdt[2], Alog[2], T3, NCH, ap3, he3);
  { int n=Bb*32; k_scan2<2><<<(n+255)/256, 256, 0, stream>>>(ap3, he3, hi3, NCH, n); }
  k_scan3<2,1024,1><<<Bb*NCH, 32, 0, stream>>>(xc3, zm3, Wx[2], Wdt[2], Bdt[2], Alog[2], Dp[2], Wo[2], hi3, T3, NCH, yout);
}